// Net_58402965291557
// MI455X (gfx1250) — compile-verified
//
#include <hip/hip_runtime.h>
#include <hip/hip_bf16.h>

typedef float v2f __attribute__((ext_vector_type(2)));
typedef float v8f __attribute__((ext_vector_type(8)));

// ---------------------------------------------------------------------------
// helpers
// ---------------------------------------------------------------------------
__device__ __forceinline__ void atomic_add_f32(float* p, float v) {
    // Force the native CDNA5 no-return f32 atomic (GLOBAL_ATOMIC_ADD_F32,
    // opcode 86, tracked on STOREcnt) — guarantees no CAS-loop fallback.
    asm volatile("global_atomic_add_f32 %0, %1, off"
                 :: "v"(p), "v"(v)
                 : "memory");
}

// ---------------------------------------------------------------------------
// zero fill (float4 stores)
// ---------------------------------------------------------------------------
__global__ void gcn_zero_kernel(float4* __restrict__ p, long long n4) {
    long long i = (long long)blockIdx.x * blockDim.x + threadIdx.x;
    if (i < n4) p[i] = make_float4(0.f, 0.f, 0.f, 0.f);
}

// ---------------------------------------------------------------------------
// degree histogram over edges (int64 indices)
// ---------------------------------------------------------------------------
__global__ void gcn_degree_kernel(const long long* __restrict__ src,
                                  const long long* __restrict__ dst,
                                  unsigned* __restrict__ deg_out,
                                  unsigned* __restrict__ deg_in,
                                  int n_edges) {
    int i = blockIdx.x * blockDim.x + threadIdx.x;
    if (i < n_edges) {
        atomicAdd(&deg_out[(int)src[i]], 1u);
        atomicAdd(&deg_in[(int)dst[i]], 1u);
    }
}

// norm = rsqrt(max(deg, 1))
__global__ void gcn_norm_kernel(const unsigned* __restrict__ deg_out,
                                const unsigned* __restrict__ deg_in,
                                float* __restrict__ norm_src,
                                float* __restrict__ norm_dst,
                                int n) {
    int i = blockIdx.x * blockDim.x + threadIdx.x;
    if (i < n) {
        norm_src[i] = rsqrtf(fmaxf((float)deg_out[i], 1.0f));
        norm_dst[i] = rsqrtf(fmaxf((float)deg_in[i], 1.0f));
    }
}

// ---------------------------------------------------------------------------
// WMMA f32 GEMM:  Out[M x NC] = (rowscale ? diag(rowscale) : I) * A[M x K] @ W[K x NC]
// One wave -> one 16x16 tile via V_WMMA_F32_16X16X4_F32; 4 waves/block = 16x64.
// Requires M % 16 == 0 (100000 = 6250*16), K % 4 == 0, NC % 64 == 0 via grid.
// ---------------------------------------------------------------------------
__global__ __launch_bounds__(128)
void gcn_gemm_wmma_kernel(const float* __restrict__ A,
                          const float* __restrict__ W,
                          float* __restrict__ Out,
                          const float* __restrict__ rowscale,  // nullptr => 1.0
                          int M, int K, int NC) {
    const int lane = threadIdx.x & 31;
    const int wave = threadIdx.x >> 5;
    const int row_base = blockIdx.x * 16;
    const int col_base = blockIdx.y * 64 + wave * 16;
    if (col_base >= NC) return;   // uniform per wave; EXEC all-1s inside

    // A layout: lane holds row m = lane&15; upper half-wave holds K+2, K+3
    const int m    = lane & 15;
    const int koff = (lane >> 4) << 1;   // 0 or 2
    const int n    = lane & 15;          // B/D column within tile

    const int row = row_base + m;
    const float s = rowscale ? rowscale[row] : 1.0f;
    const float* __restrict__ arow = A + (size_t)row * K;

    v8f acc = {};
    #pragma unroll 4
    for (int k0 = 0; k0 < K; k0 += 4) {
        const int k = k0 + koff;
        v2f a;
        a.x = arow[k] * s;
        a.y = arow[k + 1] * s;
        v2f b;
        b.x = W[(size_t)k * NC + col_base + n];
        b.y = W[(size_t)(k + 1) * NC + col_base + n];
        acc = __builtin_amdgcn_wmma_f32_16x16x4_f32(
            /*neg_a=*/false, a, /*neg_b=*/false, b,
            /*c_mod=*/(short)0, acc, /*reuse_a=*/false, /*reuse_b=*/false);
    }

    // D layout: element v -> M = v + 8*(lane>>4), N = lane&15
    const int moff = (lane >> 4) << 3;   // 0 or 8
    float* __restrict__ orow = Out + (size_t)row_base * NC + col_base;
    #pragma unroll
    for (int v = 0; v < 8; ++v)
        orow[(size_t)(v + moff) * NC + n] = acc[v];
}

// ---------------------------------------------------------------------------
// edge scatter-add: agg[dst[e], :] += h[src[e], :]   (F floats per edge)
// One wave per edge; F/32 floats per lane; native HW f32 atomics.
// ---------------------------------------------------------------------------
template <int F>
__global__ __launch_bounds__(256)
void gcn_scatter_add_kernel(const float* __restrict__ h,
                            float* __restrict__ agg,
                            const long long* __restrict__ src,
                            const long long* __restrict__ dst,
                            int n_edges) {
    const long long gid = (long long)blockIdx.x * blockDim.x + threadIdx.x;
    const int e = (int)(gid >> 5);
    if (e >= n_edges) return;
    const int lane = threadIdx.x & 31;
    const int s = (int)src[e];
    const int d = (int)dst[e];

    constexpr int PER = F / 32;   // 4 (F=128) or 2 (F=64)
    const float* __restrict__ hp = h + (size_t)s * F + lane * PER;
    float* __restrict__ ap = agg + (size_t)d * F + lane * PER;

    if constexpr (PER == 4) {
        const float4 v = *(const float4*)hp;
        atomic_add_f32(ap + 0, v.x);
        atomic_add_f32(ap + 1, v.y);
        atomic_add_f32(ap + 2, v.z);
        atomic_add_f32(ap + 3, v.w);
    } else {
        const float2 v = *(const float2*)hp;
        atomic_add_f32(ap + 0, v.x);
        atomic_add_f32(ap + 1, v.y);
    }
}

// ---------------------------------------------------------------------------
// finalize layer0: h1in = relu(agg * norm_dst + b0) * norm_src    (F = 128)
// ---------------------------------------------------------------------------
__global__ void gcn_finalize0_kernel(const float* __restrict__ agg,
                                     const float* __restrict__ norm_dst,
                                     const float* __restrict__ norm_src,
                                     const float* __restrict__ bias,
                                     float* __restrict__ out,
                                     int n_elems) {
    int idx = blockIdx.x * blockDim.x + threadIdx.x;
    if (idx < n_elems) {
        const int i = idx >> 7;       // /128
        const int f = idx & 127;
        float v = fmaf(agg[idx], norm_dst[i], bias[f]);
        v = fmaxf(v, 0.0f);
        out[idx] = v * norm_src[i];
    }
}

// finalize layer1: out = agg * norm_dst + b1    (F = 64)
__global__ void gcn_finalize1_kernel(const float* __restrict__ agg,
                                     const float* __restrict__ norm_dst,
                                     const float* __restrict__ bias,
                                     float* __restrict__ out,
                                     int n_elems) {
    int idx = blockIdx.x * blockDim.x + threadIdx.x;
    if (idx < n_elems) {
        const int i = idx >> 6;       // /64
        const int f = idx & 63;
        out[idx] = fmaf(agg[idx], norm_dst[i], bias[f]);
    }
}

// ---------------------------------------------------------------------------
// launch
// ---------------------------------------------------------------------------
extern "C" void kernel_launch(void* const* d_in, const int* in_sizes, int n_in,
                              void* d_out, int out_size, void* d_ws, size_t ws_size,
                              hipStream_t stream) {
    const int IN_F = 128, HID_F = 128, OUT_F = 64;
    const int N = in_sizes[0] / IN_F;          // 100000 (multiple of 16)
    const int E = in_sizes[1] / 2;             // 3200000

    const float*     x   = (const float*)d_in[0];
    const long long* ei  = (const long long*)d_in[1];
    const long long* src = ei;
    const long long* dst = ei + E;
    const float*     W0  = (const float*)d_in[2];
    const float*     b0  = (const float*)d_in[3];
    const float*     W1  = (const float*)d_in[4];
    const float*     b1  = (const float*)d_in[5];
    float*           out = (float*)d_out;

    // workspace carve: 2N u32 + 2N f32 + 2 * N*128 f32  (~104 MB, L2-resident)
    unsigned* deg_out  = (unsigned*)d_ws;
    unsigned* deg_in   = deg_out + N;
    float*    norm_src = (float*)(deg_in + N);
    float*    norm_dst = norm_src + N;
    float*    bufA     = norm_dst + N;            // h0 / h1in / agg1
    float*    bufB     = bufA + (size_t)N * 128;  // agg0 / h1

    const int T = 256;

    // 1) zero degree counters (2N u32)
    {
        long long n4 = ((long long)2 * N * sizeof(unsigned)) / 16;
        gcn_zero_kernel<<<(unsigned)((n4 + T - 1) / T), T, 0, stream>>>((float4*)deg_out, n4);
    }
    // 2) degree histogram
    gcn_degree_kernel<<<(E + T - 1) / T, T, 0, stream>>>(src, dst, deg_out, deg_in, E);
    // 3) norms
    gcn_norm_kernel<<<(N + T - 1) / T, T, 0, stream>>>(deg_out, deg_in, norm_src, norm_dst, N);

    // 4) GEMM0: bufA = diag(norm_src) * x @ W0   [N x 128]
    gcn_gemm_wmma_kernel<<<dim3(N / 16, HID_F / 64), 128, 0, stream>>>(
        x, W0, bufA, norm_src, N, IN_F, HID_F);

    // 5) zero agg0 (bufB, N*128 f32)
    {
        long long n4 = ((long long)N * 128) / 4;
        gcn_zero_kernel<<<(unsigned)((n4 + T - 1) / T), T, 0, stream>>>((float4*)bufB, n4);
    }
    // 6) scatter layer0: bufB[dst] += bufA[src]   (128 f/edge, 1 wave/edge)
    {
        long long nthreads = (long long)E * 32;
        gcn_scatter_add_kernel<128><<<(unsigned)((nthreads + T - 1) / T), T, 0, stream>>>(
            bufA, bufB, src, dst, E);
    }
    // 7) bufA = relu(bufB * norm_dst + b0) * norm_src
    {
        int n = N * HID_F;
        gcn_finalize0_kernel<<<(n + T - 1) / T, T, 0, stream>>>(
            bufB, norm_dst, norm_src, b0, bufA, n);
    }

    // 8) GEMM1: bufB[0 : N*64] = bufA @ W1   [N x 64]
    gcn_gemm_wmma_kernel<<<dim3(N / 16, OUT_F / 64), 128, 0, stream>>>(
        bufA, W1, bufB, nullptr, N, HID_F, OUT_F);

    // 9) zero agg1 (bufA, N*64 f32)
    {
        long long n4 = ((long long)N * 64) / 4;
        gcn_zero_kernel<<<(unsigned)((n4 + T - 1) / T), T, 0, stream>>>((float4*)bufA, n4);
    }
    // 10) scatter layer1: bufA[dst] += bufB[src]  (64 f/edge)
    {
        long long nthreads = (long long)E * 32;
        gcn_scatter_add_kernel<64><<<(unsigned)((nthreads + T - 1) / T), T, 0, stream>>>(
            bufB, bufA, src, dst, E);
    }
    // 11) out = bufA * norm_dst + b1
    {
        int n = N * OUT_F;
        gcn_finalize1_kernel<<<(n + T - 1) / T, T, 0, stream>>>(
            bufA, norm_dst, b1, out, n);
    }
}